// GraphConvolution_28759101014305
// MI455X (gfx1250) — compile-verified
//
#include <hip/hip_runtime.h>
#include <hip/hip_bf16.h>

// GCN layer: out = segment_sum(adj_vals * (input @ W)[col_idx], row_idx) + bias
// Phase 1: support = input @ W via V_WMMA_F32_16X16X4_F32 (fp32-exact)
// Phase 2: out[i][:] = bias
// Phase 3: edge scatter with hardware global_atomic_add_f32

typedef float v2f __attribute__((ext_vector_type(2)));
typedef float v8f __attribute__((ext_vector_type(8)));

#define D_FEAT 256  // D_IN == D_OUT == 256

// ---------------------------------------------------------------------------
// Kernel 1: dense GEMM  S[M,256] = A[M,256] * W[256,256]
// One wave computes one 16x16 output tile; K marched in steps of 4 with
// V_WMMA_F32_16X16X4_F32.
//
// f32 WMMA VGPR layouts (CDNA5 ISA 7.12.2):
//   A 16x4 :  lane L -> row m = L&15 ;  VGPR v holds K = (L>>4)*2 + v
//   B 4x16 :  lane L -> col n = L&15 ;  VGPR v holds K = (L>>4)*2 + v
//   C/D 16x16: lane L -> col n = L&15 ; VGPR v holds row m = v + 8*(L>>4)
// ---------------------------------------------------------------------------
__global__ __launch_bounds__(256) void gcn_gemm_wmma(
    const float* __restrict__ A,   // [M,256] row-major
    const float* __restrict__ W,   // [256,256] row-major
    float* __restrict__ S,         // [M,256] row-major
    int tilesM)                    // M/16
{
  const int lane   = threadIdx.x & 31;
  const int waveId = threadIdx.x >> 5;
  const int tile   = blockIdx.x * 8 + waveId;       // tile index over M x N tiles
  const int nTiles = tilesM * 16;                   // 16 N-tiles (256/16)
  if (tile >= nTiles) return;                       // wave-uniform exit

  const int tM = tile >> 4;      // 0 .. tilesM-1
  const int tN = tile & 15;      // 0 .. 15

  const int half  = lane >> 4;   // 0: K pair {0,1}, 1: K pair {2,3}
  const int l15   = lane & 15;
  const int m     = tM * 16 + l15;     // A row this lane feeds
  const int n     = tN * 16 + l15;     // B/C/D column this lane feeds

  // Base pointers: lane's A row offset by its K-pair; lane's W column.
  const float* arow = A + (size_t)m * D_FEAT + half * 2;
  const float* bcol = W + (size_t)(half * 2) * D_FEAT + n;

  v8f acc = {};
  #pragma unroll 8
  for (int k = 0; k < D_FEAT; k += 4) {
    // A[m][k + half*2 + {0,1}] : contiguous, 8B-aligned pair
    v2f a = *(const v2f*)(arow + k);
    // W[k + half*2 + {0,1}][n] : column walk, stride 256 floats
    v2f b;
    b.x = bcol[(size_t)k * D_FEAT];
    b.y = bcol[(size_t)k * D_FEAT + D_FEAT];
    acc = __builtin_amdgcn_wmma_f32_16x16x4_f32(
        /*neg_a=*/false, a, /*neg_b=*/false, b,
        /*c_mod=*/(short)0, acc, /*reuse_a=*/false, /*reuse_b=*/false);
  }

  // Scatter C/D layout back to row-major S.
  float* srow = S + (size_t)(tM * 16 + half * 8) * D_FEAT + tN * 16 + l15;
  #pragma unroll
  for (int v = 0; v < 8; ++v) {
    srow[(size_t)v * D_FEAT] = acc[v];
  }
}

// ---------------------------------------------------------------------------
// Kernel 2: out[i][j] = bias[j]   (seed accumulator; reference adds bias last)
// ---------------------------------------------------------------------------
__global__ __launch_bounds__(256) void gcn_bias_init(
    float* __restrict__ out, const float* __restrict__ bias, int total)
{
  int idx = blockIdx.x * 256 + threadIdx.x;
  if (idx < total) out[idx] = bias[idx & (D_FEAT - 1)];
}

// ---------------------------------------------------------------------------
// Kernel 3: COO SpMM scatter. One wave per edge; lane L covers float4 chunks
// L and L+32 of the 256-float row (2 x float4 loads -> 8 atomic f32 adds).
// ---------------------------------------------------------------------------
__global__ __launch_bounds__(256) void gcn_edge_scatter(
    const float* __restrict__ S,
    const float* __restrict__ vals,
    const int*   __restrict__ rows,
    const int*   __restrict__ cols,
    float* __restrict__ out,
    int nEdges)
{
  const int edge = (blockIdx.x * 256 + threadIdx.x) >> 5;
  const int lane = threadIdx.x & 31;
  if (edge >= nEdges) return;

  const float v = vals[edge];            // wave-uniform
  const int   c = cols[edge];
  const int   r = rows[edge];

  const float4* src = (const float4*)(S + (size_t)c * D_FEAT);
  float4 x0 = src[lane];                 // floats [lane*4, lane*4+4)
  float4 x1 = src[lane + 32];            // floats [128+lane*4, ...)

  float* dst = out + (size_t)r * D_FEAT;
  int o0 = lane * 4;
  int o1 = 128 + lane * 4;
  unsafeAtomicAdd(dst + o0 + 0, v * x0.x);
  unsafeAtomicAdd(dst + o0 + 1, v * x0.y);
  unsafeAtomicAdd(dst + o0 + 2, v * x0.z);
  unsafeAtomicAdd(dst + o0 + 3, v * x0.w);
  unsafeAtomicAdd(dst + o1 + 0, v * x1.x);
  unsafeAtomicAdd(dst + o1 + 1, v * x1.y);
  unsafeAtomicAdd(dst + o1 + 2, v * x1.z);
  unsafeAtomicAdd(dst + o1 + 3, v * x1.w);
}

// ---------------------------------------------------------------------------
extern "C" void kernel_launch(void* const* d_in, const int* in_sizes, int n_in,
                              void* d_out, int out_size, void* d_ws, size_t ws_size,
                              hipStream_t stream) {
  const float* input  = (const float*)d_in[0];  // [N, 256]
  const float* weight = (const float*)d_in[1];  // [256, 256]
  const float* bias   = (const float*)d_in[2];  // [256]
  const float* vals   = (const float*)d_in[3];  // [E]
  const int*   rows   = (const int*)d_in[4];    // [E]
  const int*   cols   = (const int*)d_in[5];    // [E]
  float* out = (float*)d_out;

  const int nNodes = in_sizes[0] / D_FEAT;      // 10000
  const int nEdges = in_sizes[3];               // 320000

  // Scratch: support matrix [nNodes, 256] fp32 (10.24 MB for N=10000).
  float* support = (float*)d_ws;

  // Phase 1: support = input @ W  (one wave per 16x16 tile, 8 waves/block)
  const int tilesM = nNodes / 16;               // 625 (10000 % 16 == 0)
  const int nTiles = tilesM * 16;               // x 16 N-tiles
  const int gemmBlocks = (nTiles + 7) / 8;
  gcn_gemm_wmma<<<gemmBlocks, 256, 0, stream>>>(input, weight, support, tilesM);

  // Phase 2: out = broadcast(bias)
  const int total = nNodes * D_FEAT;
  gcn_bias_init<<<(total + 255) / 256, 256, 0, stream>>>(out, bias, total);

  // Phase 3: scatter-add edges (8 edges per 256-thread block)
  const int edgeBlocks = (nEdges + 7) / 8;
  gcn_edge_scatter<<<edgeBlocks, 256, 0, stream>>>(support, vals, rows, cols, out, nEdges);
}